// GCN_41360535061060
// MI455X (gfx1250) — compile-verified
//
#include <hip/hip_runtime.h>

typedef __attribute__((ext_vector_type(2))) float v2f;
typedef __attribute__((ext_vector_type(8))) float v8f;

#define IN_SIZE  128
#define OUT_SIZE 64
#define EPT      8   // edges per thread in the scatter kernel

// ---------------------------------------------------------------------------
// Zero the scatter accumulator and the per-node edge counts.
// ---------------------------------------------------------------------------
__global__ __launch_bounds__(256) void gcn_zero(float* __restrict__ acc,
                                                float* __restrict__ counts,
                                                int n_nodes) {
  int t = blockIdx.x * blockDim.x + threadIdx.x;
  int total = n_nodes * OUT_SIZE;
  if (t < total)  acc[t] = 0.0f;
  if (t < n_nodes) counts[t] = 0.0f;
}

// ---------------------------------------------------------------------------
// counts[t] = number of edges whose target is t (as float).
// ---------------------------------------------------------------------------
__global__ __launch_bounds__(256) void gcn_count(const int* __restrict__ tgt,
                                                 float* __restrict__ counts,
                                                 int n_edges) {
  int e = blockIdx.x * blockDim.x + threadIdx.x;
  if (e < n_edges) atomicAdd(&counts[tgt[e]], 1.0f);
}

// ---------------------------------------------------------------------------
// H0 = X @ W + b   using V_WMMA_F32_16X16X4_F32.
// One wave32 computes one 16x16 tile of the [n_nodes x 64] output.
//   A (16x4 f32):  lane&15 = M row,  K = (lane>>4)*2 + vgpr   -> v2f per step
//   B (4x16 f32):  lane&15 = N col,  K = (lane>>4)*2 + vgpr   -> v2f per step
//   C/D (16x16):   8 VGPRs, M = v + (lane>>4)*8, N = lane&15
// ---------------------------------------------------------------------------
__global__ __launch_bounds__(256) void gcn_gemm_wmma(const float* __restrict__ X,
                                                     const float* __restrict__ W,
                                                     const float* __restrict__ bias,
                                                     float* __restrict__ out,
                                                     int n_nodes) {
  int gid  = blockIdx.x * blockDim.x + threadIdx.x;
  int wave = gid >> 5;                 // global wave id
  int lane = threadIdx.x & 31;
  int col_tile = wave & 3;             // 64 cols -> 4 tiles of 16
  int row0     = (wave >> 2) * 16;
  if (row0 >= n_nodes) return;         // wave-uniform exit (EXEC stays all-1s for WMMA)

  int r   = lane & 15;
  int kh  = lane >> 4;                 // half-wave selector for K
  int row = row0 + r;
  if (row >= n_nodes) row = n_nodes - 1;   // clamp loads; stores are guarded
  int col = col_tile * 16 + r;

  const float* xrow = X + (size_t)row * IN_SIZE;
  v8f acc = {};
#pragma unroll
  for (int kk = 0; kk < IN_SIZE / 4; ++kk) {
    int k0 = kk * 4 + kh * 2;
    v2f a, b;
    a.x = xrow[k0];
    a.y = xrow[k0 + 1];
    b.x = W[(size_t)k0       * OUT_SIZE + col];
    b.y = W[(size_t)(k0 + 1) * OUT_SIZE + col];
    acc = __builtin_amdgcn_wmma_f32_16x16x4_f32(
        /*neg_a=*/false, a, /*neg_b=*/false, b,
        /*c_mod=*/(short)0, acc, /*reuse_a=*/false, /*reuse_b=*/false);
  }

  float bv = bias[col];
#pragma unroll
  for (int v = 0; v < 8; ++v) {
    int m = row0 + v + kh * 8;
    if (m < n_nodes)
      out[(size_t)m * OUT_SIZE + col] = acc[v] + bv;
  }
}

// ---------------------------------------------------------------------------
// acc[tgt] += h[src] for every edge. tgt_idx is sorted, so each thread owns
// one feature lane over a strip of EPT edges and accumulates locally while
// the target is unchanged, flushing with a single f32 atomic per run.
// Thread layout: t = strip*64 + f  (f contiguous -> coalesced gathers/atomics)
// ---------------------------------------------------------------------------
__global__ __launch_bounds__(256) void gcn_scatter(const int* __restrict__ src_idx,
                                                   const int* __restrict__ tgt_idx,
                                                   const float* __restrict__ h,
                                                   float* __restrict__ acc,
                                                   int n_edges) {
  int t  = blockIdx.x * blockDim.x + threadIdx.x;
  int f  = t & (OUT_SIZE - 1);
  int e0 = (t >> 6) * EPT;
  if (e0 >= n_edges) return;
  int e1 = e0 + EPT;
  if (e1 > n_edges) e1 = n_edges;

  int   cur = tgt_idx[e0];
  float sum = 0.0f;
  for (int e = e0; e < e1; ++e) {
    int te = tgt_idx[e];
    if (te != cur) {
      atomicAdd(&acc[(size_t)cur * OUT_SIZE + f], sum);
      cur = te;
      sum = 0.0f;
    }
    sum += h[(size_t)src_idx[e] * OUT_SIZE + f];
  }
  atomicAdd(&acc[(size_t)cur * OUT_SIZE + f], sum);
}

// ---------------------------------------------------------------------------
// hout[t] = (m>0) ? (hin[t] + acc[t]) / (m+1) : 0   (+ optional ReLU)
// Also re-zeroes acc so the next depth's scatter starts clean.
// ---------------------------------------------------------------------------
__global__ __launch_bounds__(256) void gcn_combine(const float* __restrict__ hin,
                                                   float* __restrict__ acc,
                                                   const float* __restrict__ counts,
                                                   float* __restrict__ hout,
                                                   int n_nodes, int do_relu) {
  int t = blockIdx.x * blockDim.x + threadIdx.x;
  if (t >= n_nodes * OUT_SIZE) return;
  int   node = t >> 6;
  float m    = counts[node];
  float v    = 0.0f;
  if (m > 0.0f) v = (hin[t] + acc[t]) / (m + 1.0f);
  if (do_relu)  v = fmaxf(v, 0.0f);
  hout[t] = v;
  acc[t]  = 0.0f;
}

// ---------------------------------------------------------------------------
// Launch sequence (all on `stream`, graph-capture safe):
//   zero -> count -> WMMA gemm -> [scatter -> combine] x2 (last combine
//   fuses ReLU and writes d_out).
// ---------------------------------------------------------------------------
extern "C" void kernel_launch(void* const* d_in, const int* in_sizes, int n_in,
                              void* d_out, int out_size, void* d_ws, size_t ws_size,
                              hipStream_t stream) {
  const float* data = (const float*)d_in[0];
  const float* W    = (const float*)d_in[1];
  const float* bias = (const float*)d_in[2];
  const int*   src  = (const int*)d_in[3];
  const int*   tgt  = (const int*)d_in[4];

  int n_nodes = in_sizes[0] / IN_SIZE;
  int n_edges = in_sizes[3];

  float* h0     = (float*)d_ws;                       // [n_nodes * 64]
  float* acc    = h0  + (size_t)n_nodes * OUT_SIZE;   // [n_nodes * 64]
  float* counts = acc + (size_t)n_nodes * OUT_SIZE;   // [n_nodes]
  float* outp   = (float*)d_out;

  int nfeat = n_nodes * OUT_SIZE;
  int blk   = 256;

  // 1) zero accumulator + counts
  gcn_zero<<<(nfeat + blk - 1) / blk, blk, 0, stream>>>(acc, counts, n_nodes);

  // 2) per-target edge counts
  gcn_count<<<(n_edges + blk - 1) / blk, blk, 0, stream>>>(tgt, counts, n_edges);

  // 3) H0 = X @ W + b  (fp32 WMMA)
  {
    int row_tiles = (n_nodes + 15) / 16;
    int waves     = row_tiles * (OUT_SIZE / 16);
    int threads   = waves * 32;
    gcn_gemm_wmma<<<(threads + blk - 1) / blk, blk, 0, stream>>>(data, W, bias, h0, n_nodes);
  }

  // 4) depth 1: scatter + combine (in place, acc re-zeroed)
  {
    int strips  = (n_edges + EPT - 1) / EPT;
    int threads = strips * OUT_SIZE;
    gcn_scatter<<<(threads + blk - 1) / blk, blk, 0, stream>>>(src, tgt, h0, acc, n_edges);
    gcn_combine<<<(nfeat + blk - 1) / blk, blk, 0, stream>>>(h0, acc, counts, h0, n_nodes, 0);
  }

  // 5) depth 2: scatter + combine with fused ReLU, write final output
  {
    int strips  = (n_edges + EPT - 1) / EPT;
    int threads = strips * OUT_SIZE;
    gcn_scatter<<<(threads + blk - 1) / blk, blk, 0, stream>>>(src, tgt, h0, acc, n_edges);
    gcn_combine<<<(nfeat + blk - 1) / blk, blk, 0, stream>>>(h0, acc, counts, outp, n_nodes, 1);
  }
}